// ParallelBellowsLayers_72610717106293
// MI455X (gfx1250) — compile-verified
//
#include <hip/hip_runtime.h>
#include <stdint.h>

// ---------------- problem constants (from reference) ----------------
#define N_TECH   2
#define N_EXP    16
#define BATCH    128
#define NGENES   20000
#define C_TOTAL  (NGENES * N_TECH)   // 40000 channels

// ---------------- tiling ----------------
#define BLOCK_C      256             // channels per block (8 wave32)
#define BATCH_CHUNK  32              // batch rows per block (grid.y = 4)

typedef unsigned int uint32x4 __attribute__((ext_vector_type(4)));
typedef int          int32x8  __attribute__((ext_vector_type(8)));
typedef int          int32x4  __attribute__((ext_vector_type(4)));

#define HAVE_TDM __has_builtin(__builtin_amdgcn_tensor_load_to_lds)

#if HAVE_TDM
// 1-D TDM copy: nelem f32 elements from global to LDS byte offset.
// D# built per CDNA5 ISA 8.3/8.4 (group0 128b, group1 256b; groups 2/3 zero).
__device__ __forceinline__ void tdm_load_1d(const void* gsrc,
                                            uint32_t lds_byte_off,
                                            uint32_t nelem_f32)
{
  uint64_t ga = (uint64_t)(uintptr_t)gsrc;

  uint32x4 g0;
  g0.x = 1u;                                                  // count=1 (valid), user mode, no gather
  g0.y = lds_byte_off;                                        // lds_addr [63:32]
  g0.z = (uint32_t)ga;                                        // global_addr[31:0]
  g0.w = (uint32_t)((ga >> 32) & 0x01FFFFFFu) | (2u << 30);   // global_addr[56:32] | type=2

  int32x8 g1;
  g1[0] = (int)(2u << 16);                                    // workgroup_mask=0, data_size=2 (4B)
  g1[1] = (int)((nelem_f32 & 0xFFFFu) << 16);                 // tensor_dim0[15:0] @ bits63:48
  g1[2] = (int)(((nelem_f32 >> 16) & 0xFFFFu)                 // tensor_dim0[31:16]
                | (1u << 16));                                // tensor_dim1 = 1
  g1[3] = (int)((nelem_f32 & 0xFFFFu) << 16);                 // tensor_dim1[31:16]=0 | tile_dim0
  g1[4] = 0;                                                  // tile_dim1=0, tile_dim2=0 (unused)
  g1[5] = (int)nelem_f32;                                     // tensor_dim0_stride[31:0]
  g1[6] = 0;
  g1[7] = 0;

  int32x4 z4 = {0, 0, 0, 0};
#if __clang_major__ >= 23
  int32x8 z8 = {0, 0, 0, 0, 0, 0, 0, 0};
  __builtin_amdgcn_tensor_load_to_lds(g0, g1, z4, z4, z8, 0); // 6-arg (clang-23 lane)
#else
  __builtin_amdgcn_tensor_load_to_lds(g0, g1, z4, z4, 0);     // 5-arg (ROCm 7.2 lane)
#endif
}
#endif

// The TDM writes LDS behind the compiler's back (LDS address travels as an
// integer inside the descriptor). Escape the LDS pointer + full memory
// clobber so alias analysis cannot fold the subsequent ds_loads to undef.
__device__ __forceinline__ void lds_publish(void* p)
{
  asm volatile("" : "+v"(p) : : "memory");
}

__global__ __launch_bounds__(BLOCK_C)
void bellows_kernel(const float* __restrict__ x,
                    const float* __restrict__ w1,
                    const float* __restrict__ b1,
                    const float* __restrict__ w2,
                    const float* __restrict__ b2,
                    float* __restrict__ out)
{
  // LDS: w1|b1|w2 tiles (256*16 f32 each) + b2 tile (256 f32) = 50176 B
  __shared__ float smem[3 * BLOCK_C * N_EXP + BLOCK_C];
  float* sW1 = smem;
  float* sB1 = smem + 1 * BLOCK_C * N_EXP;
  float* sW2 = smem + 2 * BLOCK_C * N_EXP;
  float* sB2 = smem + 3 * BLOCK_C * N_EXP;

  const int tid   = threadIdx.x;
  const int cBase = blockIdx.x * BLOCK_C;
  const int nCh   = (C_TOTAL - cBase) < BLOCK_C ? (C_TOTAL - cBase) : BLOCK_C;
  const int c     = cBase + tid;

#if HAVE_TDM
  // One wave issues 4 TDM descriptors (weights are contiguous per channel tile).
  if (tid < 32) {
    tdm_load_1d(w1 + (size_t)cBase * N_EXP, 0u,                        (uint32_t)(nCh * N_EXP));
    tdm_load_1d(b1 + (size_t)cBase * N_EXP, 1u * BLOCK_C * N_EXP * 4u, (uint32_t)(nCh * N_EXP));
    tdm_load_1d(w2 + (size_t)cBase * N_EXP, 2u * BLOCK_C * N_EXP * 4u, (uint32_t)(nCh * N_EXP));
    tdm_load_1d(b2 + cBase,                 3u * BLOCK_C * N_EXP * 4u, (uint32_t)nCh);
#if __has_builtin(__builtin_amdgcn_s_wait_tensorcnt)
    __builtin_amdgcn_s_wait_tensorcnt(0);
#else
    asm volatile("s_wait_tensorcnt 0" ::: "memory");
#endif
  }
#else
  // Fallback: cooperative vectorized staging (still single-pass, coalesced).
  {
    const int n = nCh * N_EXP;
    for (int i = tid * 4; i < n; i += BLOCK_C * 4) {
      *(float4*)&sW1[i] = *(const float4*)&w1[(size_t)cBase * N_EXP + i];
      *(float4*)&sB1[i] = *(const float4*)&b1[(size_t)cBase * N_EXP + i];
      *(float4*)&sW2[i] = *(const float4*)&w2[(size_t)cBase * N_EXP + i];
    }
    if (tid < nCh) sB2[tid] = b2[cBase + tid];
  }
#endif
  __syncthreads();
  lds_publish(smem);   // make the DMA-written LDS contents visible to the optimizer

  if (tid < nCh) {
    // Register-cache this channel's private MLP (reused across 32 batch rows).
    float w1r[N_EXP], b1r[N_EXP], w2r[N_EXP];
#pragma unroll
    for (int e = 0; e < N_EXP; e += 4) {
      *(float4*)&w1r[e] = *(const float4*)&sW1[tid * N_EXP + e];
      *(float4*)&b1r[e] = *(const float4*)&sB1[tid * N_EXP + e];
      *(float4*)&w2r[e] = *(const float4*)&sW2[tid * N_EXP + e];
    }
    const float b2r = sB2[tid];

    const size_t stride = (size_t)C_TOTAL;
    const int    b0     = blockIdx.y * BATCH_CHUNK;
    const float* xp = x   + (size_t)b0 * stride + c;   // coalesced: lane == channel
    float*       op = out + (size_t)b0 * stride + c;   // out[b*C + c] == out_ref[c,b]

#pragma unroll 2
    for (int b = 0; b < BATCH_CHUNK; ++b) {
      __builtin_prefetch(xp + 8 * stride, 0, 0);       // global_prefetch_b8, stream-ahead
      const float xv = __builtin_nontemporal_load(xp); // x read exactly once -> NT
      float a0 = 0.0f, a1 = 0.0f;
#pragma unroll
      for (int e = 0; e < N_EXP; e += 2) {
        float h0 = fmaxf(fmaf(xv, w1r[e],     b1r[e]),     0.0f);
        float h1 = fmaxf(fmaf(xv, w1r[e + 1], b1r[e + 1]), 0.0f);
        a0 = fmaf(h0, w2r[e],     a0);
        a1 = fmaf(h1, w2r[e + 1], a1);
      }
      const float r = fmaxf(a0 + a1 + b2r, 0.0f);
      __builtin_nontemporal_store(r, op);              // out written exactly once -> NT
      xp += stride;
      op += stride;
    }
  }
}

extern "C" void kernel_launch(void* const* d_in, const int* in_sizes, int n_in,
                              void* d_out, int out_size, void* d_ws, size_t ws_size,
                              hipStream_t stream)
{
  const float* x  = (const float*)d_in[0];
  const float* w1 = (const float*)d_in[1];
  const float* b1 = (const float*)d_in[2];
  const float* w2 = (const float*)d_in[3];
  const float* b2 = (const float*)d_in[4];
  float* out = (float*)d_out;

  dim3 grid((C_TOTAL + BLOCK_C - 1) / BLOCK_C, BATCH / BATCH_CHUNK);
  bellows_kernel<<<grid, BLOCK_C, 0, stream>>>(x, w1, b1, w2, b2, out);
}